// out_post_Rnet_27178553049683
// MI455X (gfx1250) — compile-verified
//
#include <hip/hip_runtime.h>

// Elementwise RNet post-processing: bandwidth-bound (AI ~0.5 FLOP/byte,
// 512 MB total traffic -> ~22 us floor at 23.3 TB/s).
// Strategy: stage stride-5 AoS arrays (rect in, out) through LDS with CDNA5
// async global<->LDS b128 copies (ASYNCcnt); direct NT b64/b128 loads for
// cls/roi; stride-5 LDS access is bank-conflict-free (gcd(5,64)=1).
// Full-tile path is fully unrolled: 256 + 64 async b128 ops per stage.

typedef __attribute__((ext_vector_type(2))) float f2;
typedef __attribute__((ext_vector_type(4))) float f4;

#define TILE 256
#define TFLOATS (TILE * 5)      // 1280 floats = 5120 B per staged tile
#define TVECS (TFLOATS / 4)     // 320 b128 copies per full tile

__device__ __forceinline__ void async_g2l_b128(unsigned ldsOff, unsigned vOff,
                                               unsigned long long sBase) {
    asm volatile("global_load_async_to_lds_b128 %0, %1, %2 th:TH_LOAD_NT"
                 :: "v"(ldsOff), "v"(vOff), "s"(sBase) : "memory");
}
__device__ __forceinline__ void async_l2g_b128(unsigned vOff, unsigned ldsOff,
                                               unsigned long long sBase) {
    asm volatile("global_store_async_from_lds_b128 %0, %1, %2 th:TH_STORE_NT"
                 :: "v"(vOff), "v"(ldsOff), "s"(sBase) : "memory");
}

__global__ __launch_bounds__(TILE) void rnet_post_kernel(
    const float* __restrict__ cls, const float* __restrict__ roi,
    const float* __restrict__ rect, const float* __restrict__ hraw,
    const float* __restrict__ wraw, float* __restrict__ out,
    int N, int total)
{
    __shared__ float s_rect[TFLOATS];
    __shared__ float s_out[TFLOATS];

    const int tid        = threadIdx.x;
    const int blockStart = (int)blockIdx.x * TILE;
    const int i          = blockStart + tid;

    int count = total - blockStart;
    if (count > TILE) count = TILE;
    const bool full  = (count == TILE);
    const int nfloat = count * 5;
    const int nvec   = nfloat >> 2;
    const int nrem   = nfloat & 3;

    // ---- async-stage rect tile: global -> LDS, b128, non-temporal ----
    {
        const unsigned long long rbase = (unsigned long long)rect;
        const unsigned gbase = (unsigned)blockStart * 20u;
        const unsigned lbase = (unsigned)(unsigned long long)(&s_rect[0]);
        const unsigned loff  = lbase + (unsigned)tid * 16u;
        const unsigned voff  = gbase + (unsigned)tid * 16u;
        if (full) {
            // statically known: 320 vectors = 1 per thread + 1 for tid<64
            async_g2l_b128(loff, voff, rbase);
            if (tid < (TVECS - TILE))
                async_g2l_b128(loff + TILE * 16u, voff + TILE * 16u, rbase);
        } else {
            for (int j = tid; j < nvec; j += TILE)
                async_g2l_b128(lbase + (unsigned)j * 16u,
                               gbase + (unsigned)j * 16u, rbase);
            if (tid < nrem)
                s_rect[nvec * 4 + tid] = rect[blockStart * 5 + nvec * 4 + tid];
        }
    }

    // ---- overlap: per-thread aligned NT loads of cls (b64) and roi (b128) ----
    f2 c = {};
    f4 r = {};
    float Wb = 0.0f, Hb = 0.0f;
    if (i < total) {
        c = __builtin_nontemporal_load((const f2*)cls + i);
        r = __builtin_nontemporal_load((const f4*)roi + i);
        int b = i / N;
        Wb = wraw[b];
        Hb = hraw[b];
    }

    asm volatile("s_wait_asynccnt 0x0" ::: "memory"); // own async copies done
    __syncthreads();                                  // all waves' copies visible

    if (i < total) {
        const int s5 = tid * 5;                 // stride 5: conflict-free banks
        const float x1 = s_rect[s5 + 0];
        const float y1 = s_rect[s5 + 1];
        const float x2 = s_rect[s5 + 2];
        const float y2 = s_rect[s5 + 3];

        const float prob = c.y;
        const float w = x2 - x1;
        const float h = y2 - y1;
        const float nx1 = x1 + r.x * w;
        const float ny1 = y1 + r.y * h;
        const float nx2 = x2 + r.z * w;
        const float ny2 = y2 + r.w * h;
        const float sw = nx2 - nx1;
        const float sh = ny2 - ny1;
        const float l  = fmaxf(sw, sh);
        const float qx1 = nx1 + sw * 0.5f - l * 0.5f;
        const float qy1 = ny1 + sh * 0.5f - l * 0.5f;
        const float qx2 = qx1 + l;
        const float qy2 = qy1 + l;
        const float cx1 = fmaxf(0.0f, qx1);
        const float cy1 = fmaxf(0.0f, qy1);
        const float cx2 = fminf(Wb, qx2);
        const float cy2 = fminf(Hb, qy2);
        const bool inv = (cx1 > cx2) || (cy1 > cy2);
        const float fx1 = inv ? fminf(cx1, cx2) : cx1;
        const float fy1 = inv ? fminf(cy1, cy2) : cy1;
        const float fx2 = inv ? fmaxf(cx1, cx2) : cx2;
        const float fy2 = inv ? fmaxf(cy1, cy2) : cy2;
        const bool keep = prob >= 0.6f;

        s_out[s5 + 0] = keep ? fx1 : 0.0f;
        s_out[s5 + 1] = keep ? fy1 : 0.0f;
        s_out[s5 + 2] = keep ? fx2 : 0.0f;
        s_out[s5 + 3] = keep ? fy2 : 0.0f;
        s_out[s5 + 4] = keep ? prob : 0.0f;
    }

    __syncthreads(); // all DS writes committed before async engine reads LDS

    // ---- async drain out tile: LDS -> global, b128, non-temporal ----
    {
        const unsigned long long obase = (unsigned long long)out;
        const unsigned gbase = (unsigned)blockStart * 20u;
        const unsigned lbase = (unsigned)(unsigned long long)(&s_out[0]);
        const unsigned voff  = gbase + (unsigned)tid * 16u;
        const unsigned loff  = lbase + (unsigned)tid * 16u;
        if (full) {
            async_l2g_b128(voff, loff, obase);
            if (tid < (TVECS - TILE))
                async_l2g_b128(voff + TILE * 16u, loff + TILE * 16u, obase);
        } else {
            for (int j = tid; j < nvec; j += TILE)
                async_l2g_b128(gbase + (unsigned)j * 16u,
                               lbase + (unsigned)j * 16u, obase);
            if (tid < nrem)
                out[blockStart * 5 + nvec * 4 + tid] = s_out[nvec * 4 + tid];
        }
    }
    // S_ENDPGM performs an implicit wait-idle (drains ASYNCcnt) before the
    // workgroup slot / LDS can be reused.
}

extern "C" void kernel_launch(void* const* d_in, const int* in_sizes, int n_in,
                              void* d_out, int out_size, void* d_ws, size_t ws_size,
                              hipStream_t stream) {
    const float* cls  = (const float*)d_in[0];
    const float* roi  = (const float*)d_in[1];
    const float* rect = (const float*)d_in[2];
    const float* hraw = (const float*)d_in[3];
    const float* wraw = (const float*)d_in[4];
    float* out = (float*)d_out;

    const int B     = in_sizes[3];            // height_raw has B elements
    const int total = in_sizes[0] / 2;        // B*N (cls is B*N*2)
    const int N     = total / B;
    const int blocks = (total + TILE - 1) / TILE; // 31250 for 8M elements

    rnet_post_kernel<<<blocks, TILE, 0, stream>>>(cls, roi, rect, hraw, wraw,
                                                  out, N, total);
}